// Model_37752762531925
// MI455X (gfx1250) — compile-verified
//
#include <hip/hip_runtime.h>
#include <math.h>

#define DIM 128
#define RELS 237
#define FEPS 1e-12f

typedef __attribute__((ext_vector_type(2))) float v2f;
typedef __attribute__((ext_vector_type(8))) float v8f;

// ---------------- prep kernels ----------------

// one wave per triple: invh[n] = 1 / max(||ent[h_ids[n]]||, eps)
__global__ void k_invnorm(const float* __restrict__ ent, const int* __restrict__ ids,
                          float* __restrict__ invn, int n) {
    int wave = (blockIdx.x * blockDim.x + threadIdx.x) >> 5;
    int lane = threadIdx.x & 31;
    if (wave >= n) return;
    const float* p = ent + (size_t)ids[wave] * DIM + lane * 4;
    float4 v = *(const float4*)p;
    float s = v.x * v.x + v.y * v.y + v.z * v.z + v.w * v.w;
    for (int o = 1; o < 32; o <<= 1) s += __shfl_xor(s, o, 32);
    if (lane == 0) invn[wave] = 1.0f / fmaxf(sqrtf(s), FEPS);
}

__global__ void k_zero(int* __restrict__ p, int n) {
    int i = blockIdx.x * blockDim.x + threadIdx.x;
    if (i < n) p[i] = 0;
}

__global__ void k_count(const int* __restrict__ r_typ, int* __restrict__ counts, int n) {
    int i = blockIdx.x * blockDim.x + threadIdx.x;
    if (i < n) atomicAdd(&counts[r_typ[i]], 1);
}

__global__ void k_scan(const int* __restrict__ counts, int* __restrict__ offsets) {
    if (blockIdx.x == 0 && threadIdx.x == 0) {
        int off = 0;
        for (int i = 0; i < RELS; i++) { offsets[i] = off; off += counts[i]; }
    }
}

__global__ void k_scatter(const int* __restrict__ r_typ, const int* __restrict__ offsets,
                          int* __restrict__ cursor, int* __restrict__ order, int n) {
    int i = blockIdx.x * blockDim.x + threadIdx.x;
    if (i < n) {
        int r = r_typ[i];
        int pos = offsets[r] + atomicAdd(&cursor[r], 1);
        order[pos] = i;
    }
}

// ---------------- main WMMA kernel ----------------
// One block per relation. R_r (128x128 f32) staged in LDS with +4 row padding
// (conflict-free b64 B-fragment loads). Each of the 4 waves processes tiles of
// 16 triples: prod[16x128] = H[16x128] x R^T via 32 K-steps x 8 N-tiles of
// v_wmma_f32_16x16x4_f32 (fp32-exact). Score from norms + dot with t.
__global__ __launch_bounds__(128) void k_transr(
    const float* __restrict__ ent, const int* __restrict__ h_ids,
    const int* __restrict__ t_ids, const float* __restrict__ invh,
    const int* __restrict__ order, const int* __restrict__ counts,
    const int* __restrict__ offsets, float* __restrict__ out)
{
    __shared__ float Rlds[DIM * 132];          // 67.5 KB < 320 KB/WGP

    const int r   = blockIdx.x;
    const int tid = threadIdx.x;

    // cooperative load: thread tid copies relation row tid (R[k][d] = ent[(r*128+k)*128+d])
    {
        const float* src = ent + ((size_t)r * DIM + tid) * DIM;
        float* dst = &Rlds[tid * 132];
        #pragma unroll
        for (int j = 0; j < DIM; j += 4) {
            float4 v = *(const float4*)(src + j);
            dst[j + 0] = v.x; dst[j + 1] = v.y; dst[j + 2] = v.z; dst[j + 3] = v.w;
        }
    }
    __syncthreads();

    const int cnt    = counts[r];
    const int base   = offsets[r];
    const int ntiles = (cnt + 15) >> 4;
    const int wave   = tid >> 5;
    const int lane   = tid & 31;
    const int l15    = lane & 15;
    const int klo    = (lane >> 4) * 2;        // K sub-offset for A/B fragments

    for (int tile = wave; tile < ntiles; tile += 4) {   // wave-uniform loop
        const int gbase = base + tile * 16;
        int rem = cnt - tile * 16; if (rem > 16) rem = 16;

        // A fragment source: triple m = lane&15 (clamped for padded rows)
        const int ma  = (l15 < rem) ? l15 : (rem - 1);
        const int na  = order[gbase + ma];
        const float* apt = ent + (size_t)h_ids[na] * DIM + klo;
        const float ainv = invh[na];

        v8f acc[8];
        #pragma unroll
        for (int nt = 0; nt < 8; nt++) { v8f z = {}; acc[nt] = z; }

        for (int kk = 0; kk < 32; kk++) {
            const int kbase = kk * 4;
            v2f a = *(const v2f*)(apt + kbase);
            a *= ainv;                                  // fold h-normalization into A
            #pragma unroll
            for (int nt = 0; nt < 8; nt++) {
                // B[k][n] = R[n][k]: lane c holds col nbase+c, rows kbase+klo{+0,+1}
                v2f b = *(const v2f*)(&Rlds[(nt * 16 + l15) * 132 + kbase + klo]);
                acc[nt] = __builtin_amdgcn_wmma_f32_16x16x4_f32(
                    false, a, false, b, (short)0, acc[nt], false, false);
            }
        }

        // Scoring. C layout: VGPR e -> row e (lanes 0-15) / row e+8 (lanes 16-31),
        // column = nt*16 + (lane&15).
        const int rowbase = (lane >> 4) * 8;
        #pragma unroll
        for (int e = 0; e < 8; e++) {
            const int row  = rowbase + e;
            const int rowc = (row < rem) ? row : (rem - 1);
            const int ng   = order[gbase + rowc];
            const float* tp = ent + (size_t)t_ids[ng] * DIM + l15;

            float s2 = 0.f, st = 0.f, tt = 0.f;
            #pragma unroll
            for (int nt = 0; nt < 8; nt++) {
                float p  = acc[nt][e];
                float tv = tp[nt * 16];
                s2 += p * p; st += p * tv; tt += tv * tv;
            }
            // reduce over the 16 lanes of this half-wave (xor masks stay in-half)
            for (int o = 1; o < 16; o <<= 1) {
                s2 += __shfl_xor(s2, o, 32);
                st += __shfl_xor(st, o, 32);
                tt += __shfl_xor(tt, o, 32);
            }
            if (l15 == 0 && row < rem) {
                float np  = fmaxf(sqrtf(s2), FEPS);
                float ntm = fmaxf(sqrtf(tt), FEPS);
                float c   = st / (np * ntm);
                // ||prodN - tN|| for unit vectors = sqrt(2 - 2*cos)
                out[ng] = sqrtf(fmaxf(2.0f - 2.0f * c, 0.0f));
            }
        }
    }
}

// ---------------- launch ----------------

extern "C" void kernel_launch(void* const* d_in, const int* in_sizes, int n_in,
                              void* d_out, int out_size, void* d_ws, size_t ws_size,
                              hipStream_t stream) {
    const float* ent   = (const float*)d_in[0];
    const int*   h_ids = (const int*)d_in[1];
    const int*   r_typ = (const int*)d_in[2];
    const int*   t_ids = (const int*)d_in[3];
    float*       out   = (float*)d_out;
    const int    n     = in_sizes[1];          // N triples

    // workspace carve-out (~131 KB): invh[n] f32 | order[n] i32 | counts/offsets/cursor
    char* ws      = (char*)d_ws;
    float* invh   = (float*)ws;
    int*   order  = (int*)(ws + (size_t)n * 4);
    int*   counts = (int*)(ws + (size_t)n * 8);
    int*   offsets = counts + 256;
    int*   cursor  = offsets + 256;

    k_invnorm<<<(n + 7) / 8, 256, 0, stream>>>(ent, h_ids, invh, n);
    k_zero<<<3, 256, 0, stream>>>(counts, 768);                 // counts+offsets+cursor
    k_count<<<(n + 255) / 256, 256, 0, stream>>>(r_typ, counts, n);
    k_scan<<<1, 32, 0, stream>>>(counts, offsets);
    k_scatter<<<(n + 255) / 256, 256, 0, stream>>>(r_typ, offsets, cursor, order, n);
    k_transr<<<RELS, 128, 0, stream>>>(ent, h_ids, t_ids, invh, order, counts, offsets, out);
}